// LabeledChamferDistance_9758165696605
// MI455X (gfx1250) — compile-verified
//
#include <hip/hip_runtime.h>
#include <hip/hip_bf16.h>

typedef __attribute__((ext_vector_type(2))) float v2f;
typedef __attribute__((ext_vector_type(8))) float v8f;

#define BETA_C  1.0f
#define GAMMA_C 1.0f
#define DELTA_C 0.0f

// Monotone float <-> uint mapping (total order matches float compare).
__device__ __forceinline__ unsigned int ford(float f) {
    unsigned int u = __float_as_uint(f);
    return (u & 0x80000000u) ? ~u : (u | 0x80000000u);
}
__device__ __forceinline__ float fdec(unsigned int u) {
    return (u & 0x80000000u) ? __uint_as_float(u & 0x7fffffffu)
                             : __uint_as_float(~u);
}

__global__ void chamfer_init_keys(unsigned long long* best, int n) {
    int i = blockIdx.x * blockDim.x + threadIdx.x;
    if (i < n) best[i] = ~0ull;
}

// One workgroup = 8 wave32s; each wave owns a 16-row strip of xyz1.
// Workgroup covers 128 rows; loops all 4096 columns in 16-wide WMMA tiles.
__global__ __launch_bounds__(256)
void chamfer_main(const float* __restrict__ xyz1,
                  const float* __restrict__ xyz2,
                  unsigned long long* __restrict__ best21,
                  float* __restrict__ dist12,
                  int* __restrict__ idx12Out,
                  int N) {
    __shared__ float s_x2[4096];
    __shared__ float s_y2[4096];
    __shared__ float s_z2[4096];
    __shared__ float s_x1[128];
    __shared__ float s_y1[128];
    __shared__ float s_z1[128];

    const int nGroups = N / 128;
    const int b  = blockIdx.x / nGroups;
    const int ig = blockIdx.x % nGroups;
    const int i0wg = ig * 128;

    // Stage full xyz2 cloud for this batch into LDS (SoA).
    const float* X2 = xyz2 + (size_t)b * N * 3;
    for (int k = threadIdx.x; k < N; k += 256) {
        s_x2[k] = X2[3 * k + 0];
        s_y2[k] = X2[3 * k + 1];
        s_z2[k] = X2[3 * k + 2];
    }
    // Stage this workgroup's 128-row xyz1 strip.
    const float* X1 = xyz1 + ((size_t)b * N + i0wg) * 3;
    if (threadIdx.x < 128) {
        int k = threadIdx.x;
        s_x1[k] = X1[3 * k + 0];
        s_y1[k] = X1[3 * k + 1];
        s_z1[k] = X1[3 * k + 2];
    }
    __syncthreads();

    const int wave = threadIdx.x >> 5;
    const int lane = threadIdx.x & 31;
    const int l    = lane & 15;
    const int hi   = lane >> 4;          // 0: lanes 0-15, 1: lanes 16-31
    const int i0   = i0wg + wave * 16;   // global row base for this wave

    // Build A matrix (16x4 f32, K=3 zero-padded to 4):
    //   lanes 0-15 : VGPR0=K0(x), VGPR1=K1(y)   for row M=l
    //   lanes 16-31: VGPR0=K2(z), VGPR1=K3(0)   for row M=l
    const int lr = wave * 16 + l;
    float ax = s_x1[lr], ay = s_y1[lr], az = s_z1[lr];
    v2f A;
    A.x = hi ? az : ax;
    A.y = hi ? 0.0f : ay;

    // |a|^2 and global-row low word for the 8 rows this lane's C VGPRs map to.
    float a2v[8];
    unsigned int rowLo[8];
#pragma unroll
    for (int g = 0; g < 8; ++g) {
        int rr = wave * 16 + g + hi * 8;
        float x = s_x1[rr], y = s_y1[rr], z = s_z1[rr];
        a2v[g]  = x * x + y * y + z * z;
        rowLo[g] = (unsigned int)(i0 + g + hi * 8);
    }

    float runV[8];
    int   runI[8];
#pragma unroll
    for (int g = 0; g < 8; ++g) { runV[g] = 3.4e38f; runI[g] = 0; }

    unsigned long long* bestB = best21 + (size_t)b * N;
    const int nTiles = N / 16;

#pragma unroll 2
    for (int t = 0; t < nTiles; ++t) {
        const int jj = t * 16 + l;   // this lane's column in the tile
        float qx = s_x2[jj], qy = s_y2[jj], qz = s_z2[jj];
        float b2l = qx * qx + qy * qy + qz * qz;

        // B matrix (4x16), same half-wave K split as A.
        v2f Bv;
        Bv.x = hi ? qz : qx;
        Bv.y = hi ? 0.0f : qy;

        v8f c = {0.f, 0.f, 0.f, 0.f, 0.f, 0.f, 0.f, 0.f};
        c = __builtin_amdgcn_wmma_f32_16x16x4_f32(
                /*neg_a=*/false, A, /*neg_b=*/false, Bv,
                /*c_mod=*/(short)0, c, /*reuse_a=*/false, /*reuse_b=*/false);

        // Branchless reductions: column-direction via packed u64 keys,
        // row-direction via strict-< register selects.
        unsigned long long colKey = ~0ull;
#pragma unroll
        for (int g = 0; g < 8; ++g) {
            // dist = |a|^2 + |b|^2 - 2 a.b
            float dist = fmaf(-2.0f, c[g], a2v[g] + b2l);

            int lt = dist < runV[g];
            runV[g] = lt ? dist : runV[g];
            runI[g] = lt ? jj   : runI[g];

            unsigned long long key =
                ((unsigned long long)ford(dist) << 32) | rowLo[g];
            colKey = (key < colKey) ? key : colKey;
        }
        // Combine the two half-wave contributions for this column.
        unsigned long long oKey = __shfl_xor(colKey, 16, 32);
        colKey = (oKey < colKey) ? oKey : colKey;

        if (!hi) {
            atomicMin(&bestB[jj], colKey);   // L2-resident u64 atomic argmin
        }
    }

    // Final row-direction reduction: min over the 16 lanes of each half,
    // as u64 keys so dist-then-index ordering is a single integer compare.
#pragma unroll
    for (int g = 0; g < 8; ++g) {
        unsigned long long key =
            ((unsigned long long)ford(runV[g]) << 32) | (unsigned int)runI[g];
#pragma unroll
        for (int m = 1; m < 16; m <<= 1) {
            unsigned long long o = __shfl_xor(key, m, 32);
            key = (o < key) ? o : key;
        }
        if (l == 0) {
            int row = (int)rowLo[g];
            dist12[(size_t)b * N + row]   = fdec((unsigned int)(key >> 32));
            idx12Out[(size_t)b * N + row] = (int)(key & 0xffffffffu);
        }
    }
}

__global__ __launch_bounds__(256)
void chamfer_finalize(const float* __restrict__ dist12,
                      const unsigned long long* __restrict__ best21,
                      float* __restrict__ out,
                      int* __restrict__ idx21Out,
                      int B, int N) {
    __shared__ float sA[256], sB[256], sC[256];
    __shared__ float s_loss;
    const int tid = threadIdx.x;
    if (tid == 0) s_loss = 0.0f;
    __syncthreads();

    for (int b = 0; b < B; ++b) {
        float sum12 = 0.0f, mx12 = -3.4e38f, sum21 = 0.0f;
        for (int k = tid; k < N; k += 256) {
            float d = dist12[(size_t)b * N + k];
            sum12 += d;
            mx12 = fmaxf(mx12, d);
            unsigned long long key = best21[(size_t)b * N + k];
            sum21 += fdec((unsigned int)(key >> 32));
            idx21Out[(size_t)b * N + k] = (int)(key & 0xffffffffu);
        }
        sA[tid] = sum12; sB[tid] = mx12; sC[tid] = sum21;
        __syncthreads();
        for (int s = 128; s > 0; s >>= 1) {
            if (tid < s) {
                sA[tid] += sA[tid + s];
                sB[tid]  = fmaxf(sB[tid], sB[tid + s]);
                sC[tid] += sC[tid + s];
            }
            __syncthreads();
        }
        if (tid == 0) {
            float invN = 1.0f / (float)N;
            float lb = sA[0] * invN + sB[0] * BETA_C +
                       (GAMMA_C + DELTA_C * (float)N) * (sC[0] * invN);
            s_loss += lb;
        }
        __syncthreads();
    }
    if (tid == 0) out[0] = s_loss / (float)B;
}

extern "C" void kernel_launch(void* const* d_in, const int* in_sizes, int n_in,
                              void* d_out, int out_size, void* d_ws, size_t ws_size,
                              hipStream_t stream) {
    const float* xyz1 = (const float*)d_in[0];
    const float* xyz2 = (const float*)d_in[1];

    const int B = 4;
    const int N = in_sizes[0] / (B * 3);   // 4096

    unsigned long long* best21 = (unsigned long long*)d_ws;
    float* dist12 = (float*)((char*)d_ws + (size_t)B * N * sizeof(unsigned long long));

    float* lossOut   = (float*)d_out;
    int*   idx12Out  = (int*)d_out + 1;
    int*   idx21Out  = (int*)d_out + 1 + B * N;

    int total = B * N;
    chamfer_init_keys<<<(total + 255) / 256, 256, 0, stream>>>(best21, total);

    dim3 grid(B * (N / 128));
    chamfer_main<<<grid, 256, 0, stream>>>(xyz1, xyz2, best21, dist12, idx12Out, N);

    chamfer_finalize<<<1, 256, 0, stream>>>(dist12, best21, lossOut, idx21Out, B, N);
}